// RegionProposalNetwork_6828998001537
// MI455X (gfx1250) — compile-verified
//
#include <hip/hip_runtime.h>
#include <hip/hip_bf16.h>

typedef __attribute__((ext_vector_type(2))) float        v2f;
typedef __attribute__((ext_vector_type(8))) float        v8f;
typedef __attribute__((ext_vector_type(4))) unsigned int v4u;
typedef __attribute__((ext_vector_type(8))) int          v8i;
typedef __attribute__((ext_vector_type(4))) int          v4i;

namespace {
constexpr int kCin      = 512;
constexpr int kHW       = 10000;   // 100*100
constexpr int kKtot     = 4608;    // 512*9
constexpr int kKc       = 36;      // 4 input channels * 9 taps per chunk
constexpr int kChunks   = 128;     // 512/4
constexpr int kNpix     = 64;      // pixels per workgroup
constexpr int kThreads  = 512;     // 16 waves
constexpr int kClsElems = 8 * 9 * kHW; // 720000 (cls block precedes bbox block in d_out)

// LDS map (floats): W0 [0,18432)  W1 [18432,36864)  B0 [36864,39168)  B1 [39168,41472)
// Feat tile (head phase) reuses [0,32768).
constexpr int kW0 = 0;
constexpr int kW1 = 18432;
constexpr int kB0 = 36864;
constexpr int kB1 = 39168;
constexpr int kLdsFloats = 41472;  // 165,888 bytes
}

__device__ __forceinline__ v8f wmma_f32(v2f a, v2f b, v8f c) {
  // D = A(16x4 f32) * B(4x16 f32) + C(16x16 f32)
  return __builtin_amdgcn_wmma_f32_16x16x4_f32(false, a, false, b, (short)0, c,
                                               false, false);
}

// TDM: DMA a 512-row x 36-float tile (row stride 4608 floats) from global to LDS.
// D# built per CDNA5 ISA ch.8: group0 = {count/type/addresses}, group1 = {dims}.
__device__ __forceinline__ void tdm_load_w(const float* gsrc, float* ldst) {
  unsigned long long ga = (unsigned long long)gsrc;
  unsigned lds_off = (unsigned)(unsigned long long)(void*)ldst; // LDS byte offset = addr[31:0]

  v4u g0 = { 1u,                                   // [1:0] count=1 (valid user D#)
             lds_off,                              // [63:32] lds_addr
             (unsigned)(ga & 0xffffffffu),         // [95:64] global_addr lo
             (unsigned)((ga >> 32) & 0x01ffffffu)  // [120:96] global_addr hi
               | (2u << 30) };                     // [127:126] type=2 ("image")
  v8i g1 = { (int)(2u << 16),        // [17:16] data_size=2 -> 4 bytes; mask/flags 0
             (int)(4608u << 16),     // [63:48] tensor_dim0[15:0] = 4608
             (int)(512u << 16),      // [79:64] dim0 hi=0; [95:80] tensor_dim1[15:0]=512
             (int)(36u << 16),       // [111:96] dim1 hi=0; [127:112] tile_dim0=36
             (int)512,               // [143:128] tile_dim1=512; tile_dim2=0
             (int)4608,              // [191:160] tensor_dim0_stride lo = 4608
             0, 0 };                 // stride hi, tensor_dim1_stride = 0
  v4i g2 = { 0, 0, 0, 0 };           // unused (2D tile)
  v4i g3 = { 0, 0, 0, 0 };
#if defined(__clang_major__) && (__clang_major__ >= 23)
  v8i g4 = { 0, 0, 0, 0, 0, 0, 0, 0 };
  __builtin_amdgcn_tensor_load_to_lds(g0, g1, g2, g3, g4, 0);
#else
  __builtin_amdgcn_tensor_load_to_lds(g0, g1, g2, g3, 0);
#endif
}

__global__ __launch_bounds__(kThreads)
void rpn_fused_kernel(const float* __restrict__ x,
                      const float* __restrict__ w_rpn,
                      const float* __restrict__ b_rpn,
                      const float* __restrict__ w_cls,
                      const float* __restrict__ b_cls,
                      const float* __restrict__ w_bbox,
                      const float* __restrict__ b_bbox,
                      float* __restrict__ out)
{
  __shared__ float lds[kLdsFloats];
  float* __restrict__ Fl = lds;  // head-phase feat tile, pair-interleaved [256][128]

  const int tid  = threadIdx.x;
  const int wave = tid >> 5;
  const int lane = tid & 31;
  const int h    = lane >> 4;   // half-wave selector (K/M offset)
  const int l    = lane & 15;   // row/col within tile

  const int pBase = blockIdx.x * kNpix;
  const int batch = blockIdx.y;
  const float* __restrict__ xb = x + (size_t)batch * kCin * kHW;

  // Accumulators: wave owns output channels [32*wave, 32*wave+32) x 64 pixels.
  v8f acc[2][4];
  #pragma unroll
  for (int mt = 0; mt < 2; ++mt)
    #pragma unroll
    for (int nt = 0; nt < 4; ++nt)
      acc[mt][nt] = (v8f){0.f,0.f,0.f,0.f,0.f,0.f,0.f,0.f};

  // Register double-buffer for the im2col staging (gather + border zeros).
  float breg[5];

  auto load_b_chunk = [&](int cc) {
    const int cbase = cc * 4;
    #pragma unroll
    for (int j = 0; j < 5; ++j) {
      int idx = j * kThreads + tid;      // 0..2303 valid
      float val = 0.f;
      if (idx < kKc * kNpix) {
        int kk  = idx >> 6;              // 0..35
        int n   = idx & 63;
        int c9  = kk / 9;
        int tap = kk - c9 * 9;
        int dy  = tap / 3 - 1;
        int dx  = tap - (tap / 3) * 3 - 1;
        int p   = pBase + n;
        if (p < kHW) {
          int py = p / 100;
          int yy = py + dy;
          int xx = p - py * 100 + dx;
          if ((unsigned)yy < 100u && (unsigned)xx < 100u)
            val = xb[(cbase + c9) * kHW + yy * 100 + xx];
        }
      }
      breg[j] = val;
    }
  };

  auto store_b_chunk = [&](float* Bl) {
    #pragma unroll
    for (int j = 0; j < 5; ++j) {
      int idx = j * kThreads + tid;
      if (idx < kKc * kNpix) {
        int kk = idx >> 6;
        int n  = idx & 63;
        Bl[(kk >> 1) * 128 + 2 * n + (kk & 1)] = breg[j];  // pair-interleave for b64 reads
      }
    }
  };

  // ---- prologue: start chunk 0 (weights via TDM, im2col via registers) ----
  if (wave == 0) tdm_load_w(w_rpn, lds + kW0);
  load_b_chunk(0);

  int buf = 0;
  for (int cc = 0; cc < kChunks; ++cc) {
    __syncthreads();                     // WAR: prior chunk's readers are done
    store_b_chunk(lds + (buf ? kB1 : kB0));
    if (cc + 1 < kChunks) {
      if (wave == 0)
        tdm_load_w(w_rpn + (cc + 1) * kKc, lds + (buf ? kW0 : kW1)); // alternate buffer
      load_b_chunk(cc + 1);              // overlaps with the WMMAs below
    }
    if (wave == 0) {
      if (cc + 1 < kChunks)
        __builtin_amdgcn_s_wait_tensorcnt(1);   // only the in-flight prefetch remains
      else
        __builtin_amdgcn_s_wait_tensorcnt(0);   // final chunk fully landed
    }
    __syncthreads();                     // RAW: W[buf] DMA + B[buf] stores visible

    const float* Wb  = lds + (buf ? kW1 : kW0);
    const float* Bb  = lds + (buf ? kB1 : kB0);
    const float* wa0 = Wb + (wave * 32 + l) * kKc + 2 * h;
    const float* wa1 = wa0 + 16 * kKc;
    #pragma unroll
    for (int k0 = 0; k0 < kKc; k0 += 4) {
      v2f a0 = *(const v2f*)(wa0 + k0);          // ds_load_b64, conflict-free
      v2f a1 = *(const v2f*)(wa1 + k0);
      #pragma unroll
      for (int nt = 0; nt < 4; ++nt) {
        v2f bv = *(const v2f*)(Bb + ((k0 >> 1) + h) * 128 + 2 * (nt * 16 + l));
        acc[0][nt] = wmma_f32(a0, bv, acc[0][nt]);
        acc[1][nt] = wmma_f32(a1, bv, acc[1][nt]);
      }
    }
    buf ^= 1;
  }

  // Epilogue of the 3x3 conv: bias + ReLU in registers.
  #pragma unroll
  for (int mt = 0; mt < 2; ++mt) {
    #pragma unroll
    for (int r = 0; r < 8; ++r) {
      float bv = b_rpn[wave * 32 + mt * 16 + 8 * h + r];
      #pragma unroll
      for (int nt = 0; nt < 4; ++nt) {
        float f = acc[mt][nt][r] + bv;
        acc[mt][nt][r] = f > 0.f ? f : 0.f;
      }
    }
  }

  // Deposit feat tile (512 x 64) into LDS, pair-interleaved for the head GEMM's B operand.
  __syncthreads();
  #pragma unroll
  for (int mt = 0; mt < 2; ++mt) {
    #pragma unroll
    for (int r = 0; r < 8; ++r) {
      int ch   = wave * 32 + mt * 16 + 8 * h + r;
      int base = (ch >> 1) * 128 + (ch & 1);
      #pragma unroll
      for (int nt = 0; nt < 4; ++nt)
        Fl[base + 2 * (nt * 16 + l)] = acc[mt][nt][r];
    }
  }
  __syncthreads();

  // Fused 1x1 heads: [45x512] @ feat[512x64], M padded to 48 => 3 Mtiles x 4 Ntiles,
  // one tile per wave (12 of 16 waves; uniform branch keeps EXEC all-ones in WMMA).
  if (wave < 12) {
    const int mtH = wave >> 2;
    const int ntH = wave & 3;
    const int m   = mtH * 16 + l;
    const float keep = (m < 45) ? 1.f : 0.f;           // zero the 3 pad rows
    const float* wh = (m < 9) ? (w_cls + m * kCin)
                              : (w_bbox + ((m < 45 ? m : 44) - 9) * kCin);
    v8f accH = (v8f){0.f,0.f,0.f,0.f,0.f,0.f,0.f,0.f};
    #pragma unroll 4
    for (int k0 = 0; k0 < kCin; k0 += 4) {
      v2f a = *(const v2f*)(wh + k0 + 2 * h);
      a *= keep;
      v2f bv = *(const v2f*)(Fl + ((k0 >> 1) + h) * 128 + 2 * (ntH * 16 + l));
      accH = wmma_f32(a, bv, accH);
    }
    #pragma unroll
    for (int r = 0; r < 8; ++r) {
      int ch = mtH * 16 + 8 * h + r;
      int p  = pBase + ntH * 16 + l;
      if (p < kHW && ch < 45) {
        if (ch < 9)
          out[(size_t)batch * 9 * kHW + ch * kHW + p] = accH[r] + b_cls[ch];
        else
          out[(size_t)kClsElems + (size_t)batch * 36 * kHW + (ch - 9) * kHW + p]
              = accH[r] + b_bbox[ch - 9];
      }
    }
  }
}

extern "C" void kernel_launch(void* const* d_in, const int* in_sizes, int n_in,
                              void* d_out, int out_size, void* d_ws, size_t ws_size,
                              hipStream_t stream) {
  const float* x      = (const float*)d_in[0];
  const float* w_rpn  = (const float*)d_in[1];
  const float* b_rpn  = (const float*)d_in[2];
  const float* w_cls  = (const float*)d_in[3];
  const float* b_cls  = (const float*)d_in[4];
  const float* w_bbox = (const float*)d_in[5];
  const float* b_bbox = (const float*)d_in[6];
  float* out = (float*)d_out;

  dim3 grid((kHW + kNpix - 1) / kNpix, 8);  // 157 pixel blocks x 8 batches
  rpn_fused_kernel<<<grid, kThreads, 0, stream>>>(x, w_rpn, b_rpn, w_cls, b_cls,
                                                  w_bbox, b_bbox, out);
}